// Upsample_3813930959349
// MI455X (gfx1250) — compile-verified
//
#include <hip/hip_runtime.h>

// Upsample/resample kernel for MI455X (gfx1250).
// out[b, 0:N]    = positions[b, 0:N]                        (async DMA copy)
// out[b, N:2N]   = positions[b, idx[b,:]] + 0.05*noise[b,:] (L2-resident gather)
//
// Memory-bound: ~640 MB compulsory HBM traffic -> ~27us floor @ 23.3 TB/s.
// NT hints on read-once/write-once streams keep the 128 MB gather-hot
// positions set resident in the 192 MB L2. The verbatim copy half uses the
// CDNA5 async global<->LDS DMA path (ASYNCcnt) via inline asm. The async
// load is issued FIRST and the asynccnt wait LAST among the loads, so the
// wave's independent gather-stream loads are all in flight before it stalls.

typedef __attribute__((ext_vector_type(4))) float v4f;
typedef __attribute__((ext_vector_type(4))) int   v4i;

#define SIGMA_F 0.05f
constexpr int kNOld   = 262144;   // points per segment
constexpr int kRatio  = 2;        // upsampling ratio
constexpr int kNExtra = kNOld * (kRatio - 1);
constexpr int kB      = 128;      // segments
constexpr int kTPB    = 256;      // 8 wave32 waves per block
constexpr int kVec    = 4;        // float4 / b128 per thread
constexpr int kElemsPerBlock = kTPB * kVec;  // 1024

__global__ __launch_bounds__(kTPB) void upsample_resample_kernel(
    const float* __restrict__ positions,
    const int*   __restrict__ idx,
    const float* __restrict__ noise,
    float*       __restrict__ out)
{
  const int    b      = blockIdx.y;
  const size_t segIn  = (size_t)b * kNOld;            // positions segment base
  const size_t segEx  = (size_t)b * kNExtra;          // idx/noise segment base
  const size_t segOut = (size_t)b * (kNOld * kRatio); // output segment base
  const int    e0     = (blockIdx.x * kTPB + (int)threadIdx.x) * kVec;

  // ---- stage 1: kick off the async global->LDS load for the copy half ----
  // Per-lane private 16B LDS slice; tracked on ASYNCcnt; no barriers needed.
  __shared__ char stage[kTPB * 16];
  const float* gsrc = positions + segIn + e0;
  float*       gdst = out + segOut + e0;
  // Generic LDS address carries the LDS byte offset in its low 32 bits
  // (ISA aperture rule).
  unsigned lds_addr =
      (unsigned)(unsigned long long)(stage + (size_t)threadIdx.x * 16);

  // load: vdst = LDS byte-address VGPR, vaddr = 64-bit global address
  asm volatile("global_load_async_to_lds_b128 %0, %1, off"
               :
               : "v"(lds_addr), "v"(gsrc)
               : "memory");

  // ---- stage 2: issue all independent gather-half loads (overlap DMA) ----
  // idx/noise are read-once -> NT loads; gather loads of positions use
  // default (RT) policy so the per-segment 1 MB slice (128 MB total) stays
  // hot in the 192 MB L2.
  const float* __restrict__ seg = positions + segIn;
  v4i id = __builtin_nontemporal_load((const v4i*)(idx   + segEx + e0));
  v4f nz = __builtin_nontemporal_load((const v4f*)(noise + segEx + e0));

  v4f g;
  g.x = seg[id.x];
  g.y = seg[id.y];
  g.z = seg[id.z];
  g.w = seg[id.w];

  // ---- stage 3: drain the async load, launch the async LDS->global store ----
  asm volatile("s_wait_asynccnt 0x0" ::: "memory");
  // store: vaddr = 64-bit global address, vdata = LDS byte-address VGPR
  asm volatile("global_store_async_from_lds_b128 %0, %1, off"
               :
               : "v"(gdst), "v"(lds_addr)
               : "memory");
  // async store drains at S_ENDPGM (implicit wait-idle)

  // ---- stage 4: FMA + write-once NT store of the resampled half ----
  v4f r;
  r.x = __builtin_fmaf(SIGMA_F, nz.x, g.x);
  r.y = __builtin_fmaf(SIGMA_F, nz.y, g.y);
  r.z = __builtin_fmaf(SIGMA_F, nz.z, g.z);
  r.w = __builtin_fmaf(SIGMA_F, nz.w, g.w);

  __builtin_nontemporal_store(r, (v4f*)(out + segOut + kNOld + e0));
}

extern "C" void kernel_launch(void* const* d_in, const int* in_sizes, int n_in,
                              void* d_out, int out_size, void* d_ws, size_t ws_size,
                              hipStream_t stream) {
  (void)in_sizes; (void)n_in; (void)out_size; (void)d_ws; (void)ws_size;
  const float* positions = (const float*)d_in[0];
  const int*   idx       = (const int*)d_in[1];
  const float* noise     = (const float*)d_in[2];
  float*       out       = (float*)d_out;

  dim3 grid(kNOld / kElemsPerBlock, kB);  // (256, 128) blocks
  upsample_resample_kernel<<<grid, kTPB, 0, stream>>>(positions, idx, noise, out);
}